// PoolingAggregator_4140348473474
// MI455X (gfx1250) — compile-verified
//
#include <hip/hip_runtime.h>
#include <hip/hip_bf16.h>

// Problem constants (match the reference file)
constexpr int B = 512;      // batch rows
constexpr int N = 10000;    // feature columns
constexpr int G = 2048;     // num segments/groups
constexpr int T = 131072;   // gathered elements

typedef __attribute__((ext_vector_type(2))) float v2f;
typedef __attribute__((ext_vector_type(8))) float v8f;

__device__ __forceinline__ v8f wmma4(v2f a, v2f b, v8f c) {
    return __builtin_amdgcn_wmma_f32_16x16x4_f32(
        false, a, false, b, (short)0, c, false, false);
}

// ---------------------------------------------------------------------------
// 1) Transpose feat (B x N, row major) -> featT (N x B) so that column gathers
//    become contiguous 64B reads per 16-lane half wave.
// ---------------------------------------------------------------------------
__global__ __launch_bounds__(256) void transpose_kernel(
    const float* __restrict__ in, float* __restrict__ outT, int rows, int cols)
{
    __shared__ float tile[32][33];
    const int c0 = blockIdx.x * 32;
    const int r0 = blockIdx.y * 32;
    for (int i = threadIdx.y; i < 32; i += blockDim.y) {
        int r = r0 + i, c = c0 + threadIdx.x;
        tile[i][threadIdx.x] = (r < rows && c < cols) ? in[(size_t)r * cols + c] : 0.0f;
    }
    __syncthreads();
    for (int i = threadIdx.y; i < 32; i += blockDim.y) {
        int c = c0 + i, r = r0 + threadIdx.x;
        if (c < cols && r < rows) outT[(size_t)c * rows + r] = tile[threadIdx.x][i];
    }
}

// ---------------------------------------------------------------------------
// 2) start[g] = lower_bound(seg, g) for g in [0, G]; counts = start[g+1]-start[g]
// ---------------------------------------------------------------------------
__global__ void bounds_kernel(const int* __restrict__ seg, int* __restrict__ start,
                              int t_total, int g_total)
{
    int g = blockIdx.x * blockDim.x + threadIdx.x;
    if (g > g_total) return;
    int lo = 0, hi = t_total;
    while (lo < hi) {
        int mid = (lo + hi) >> 1;
        if (seg[mid] < g) lo = mid + 1; else hi = mid;
    }
    start[g] = lo;
}

// ---------------------------------------------------------------------------
// 3) Segmented mean via V_WMMA_F32_16X16X4_F32 with a one-hot B operand.
//    Block = 4 waves; one g-tile (16 groups) per block. Each wave owns TWO
//    16-row b-tiles (32 rows) and feeds the same one-hot into two WMMAs,
//    halving per-WMMA index/one-hot overhead. idx/seg staged through LDS.
// ---------------------------------------------------------------------------
template <bool TRANS>
__global__ __launch_bounds__(128) void segmean_wmma_kernel(
    const float* __restrict__ feat,   // TRANS: featT (N x B); else feat (B x N)
    const int*   __restrict__ idx,
    const int*   __restrict__ seg,
    const int*   __restrict__ start,
    float*       __restrict__ out)    // (B x G)
{
    constexpr int CHUNK = 256;
    __shared__ int sIdx[CHUNK];
    __shared__ int sSeg[CHUNK];

    const int lane = threadIdx.x & 31;
    const int wave = threadIdx.x >> 5;
    const int g0 = blockIdx.x * 16;
    const int b0 = blockIdx.y * 128 + wave * 32;   // wave owns rows [b0, b0+32)

    const int tStart = start[g0];
    const int tEnd   = start[g0 + 16];             // start has G+1 entries

    // ISA A-layout (32-bit A 16x4): lanes 0-15 hold K=0,1; lanes 16-31 hold K=2,3
    const int kk   = (lane & 16) ? 2 : 0;
    const int m    = lane & 15;                    // M row (A) / N col (B,C,D)
    const int gsel = g0 + m;

    const float* fr0 = TRANS ? (feat + (b0 + m))
                             : (feat + (size_t)(b0 + m) * N);
    const float* fr1 = TRANS ? (feat + (b0 + 16 + m))
                             : (feat + (size_t)(b0 + 16 + m) * N);

    v8f acc0 = {};
    v8f acc1 = {};

    // One 4-deep k-step: 2 WMMAs sharing one one-hot operand.
    auto step = [&](int id0, int id1, int sg0, int sg1) {
        v2f a0, a1, bm;
        a0.x = TRANS ? fr0[id0 * B] : fr0[id0];
        a0.y = TRANS ? fr0[id1 * B] : fr0[id1];
        a1.x = TRANS ? fr1[id0 * B] : fr1[id0];
        a1.y = TRANS ? fr1[id1 * B] : fr1[id1];
        bm.x = (sg0 == gsel) ? 1.0f : 0.0f;
        bm.y = (sg1 == gsel) ? 1.0f : 0.0f;
        acc0 = wmma4(a0, bm, acc0);
        acc1 = wmma4(a1, bm, acc1);
    };

    for (int t = tStart; t < tEnd; t += CHUNK) {
        int nv = tEnd - t;
        if (nv > CHUNK) nv = CHUNK;

        __syncthreads();   // previous slab fully consumed
        for (int i = threadIdx.x; i < nv; i += blockDim.x) {
            sIdx[i] = idx[t + i];
            sSeg[i] = seg[t + i];
        }
        // warm next slab of index data into cache
        if (t + CHUNK + (int)threadIdx.x < tEnd) {
            __builtin_prefetch(&idx[t + CHUNK + threadIdx.x], 0, 0);
            __builtin_prefetch(&seg[t + CHUNK + threadIdx.x], 0, 0);
        }
        __syncthreads();

        const int nb = nv & ~3;            // guard-free bulk
#pragma unroll 4
        for (int k = 0; k < nb; k += 4) {
            const int i0 = k + kk;
            step(sIdx[i0], sIdx[i0 + 1], sSeg[i0], sSeg[i0 + 1]);
        }
        if (nb < nv) {                     // uniform masked tail (last slab only)
            const int i0 = nb + kk, i1 = i0 + 1;
            const int id0 = (i0 < nv) ? sIdx[i0] : 0;
            const int id1 = (i1 < nv) ? sIdx[i1] : 0;
            const int sg0 = (i0 < nv) ? sSeg[i0] : -1;
            const int sg1 = (i1 < nv) ? sSeg[i1] : -1;
            step(id0, id1, sg0, sg1);
        }
    }

    // Epilogue: divide column by count, store (streamed, never re-read).
    // C/D layout: VGPR r -> M=r (lanes 0-15) / M=r+8 (lanes 16-31); N = lane&15.
    const float cnt = (float)(start[gsel + 1] - start[gsel]);
    const float inv = 1.0f / fmaxf(cnt, 1.0f);
    const int rbase = b0 + ((lane & 16) ? 8 : 0);
#pragma unroll
    for (int r = 0; r < 8; ++r) {
        __builtin_nontemporal_store(acc0[r] * inv,
            &out[(size_t)(rbase + r) * G + gsel]);
        __builtin_nontemporal_store(acc1[r] * inv,
            &out[(size_t)(rbase + 16 + r) * G + gsel]);
    }
}

// ---------------------------------------------------------------------------
extern "C" void kernel_launch(void* const* d_in, const int* in_sizes, int n_in,
                              void* d_out, int out_size, void* d_ws, size_t ws_size,
                              hipStream_t stream)
{
    (void)in_sizes; (void)n_in; (void)out_size;

    const float* feat = (const float*)d_in[0];
    const int*   idx  = (const int*)d_in[1];
    const int*   seg  = (const int*)d_in[2];
    float*       out  = (float*)d_out;

    const size_t featT_bytes = (size_t)N * B * sizeof(float);
    const size_t need = featT_bytes + (size_t)(G + 1) * sizeof(int);

    if (ws_size >= need) {
        float* featT = (float*)d_ws;
        int*   start = (int*)((char*)d_ws + featT_bytes);

        dim3 tb(32, 8);
        dim3 tg((N + 31) / 32, (B + 31) / 32);
        transpose_kernel<<<tg, tb, 0, stream>>>(feat, featT, B, N);

        bounds_kernel<<<(G + 1 + 255) / 256, 256, 0, stream>>>(seg, start, T, G);

        dim3 grid(G / 16, B / 128);
        segmean_wmma_kernel<true><<<grid, 128, 0, stream>>>(featT, idx, seg, start, out);
    } else {
        // Workspace too small for the transpose: strided gather fallback.
        int* start = (int*)d_ws;   // needs (G+1)*4 = 8196 bytes
        bounds_kernel<<<(G + 1 + 255) / 256, 256, 0, stream>>>(seg, start, T, G);

        dim3 grid(G / 16, B / 128);
        segmean_wmma_kernel<false><<<grid, 128, 0, stream>>>(feat, idx, seg, start, out);
    }
}